// GCN_29712583753989
// MI455X (gfx1250) — compile-verified
//
#include <hip/hip_runtime.h>

// Problem constants (match reference)
#define NB   8
#define CBGN 30000
#define POIN 20000
#define NE   1600000
#define NN   50000
// F_IN=8, F_HID=16, F_OUT=8

typedef __attribute__((ext_vector_type(2))) float v2f;
typedef __attribute__((ext_vector_type(8))) float v8f;

// ---------------------------------------------------------------- utilities
__global__ void zero_f32(float* __restrict__ p, long long n) {
  long long i = (long long)blockIdx.x * blockDim.x + threadIdx.x;
  if (i < n) p[i] = 0.0f;
}

// Structural degrees (batch-independent): deg_out from src, deg_in from dst.
__global__ void degree_kernel(const int* __restrict__ src, const int* __restrict__ dst,
                              float* __restrict__ degO, float* __restrict__ degI) {
  int e = blockIdx.x * blockDim.x + threadIdx.x;
  if (e >= NE) return;
  atomicAdd(&degO[src[e]], 1.0f);
  atomicAdd(&degI[dst[e]], 1.0f);
}

// In-place: deg -> clip(deg,1)^-0.5
__global__ void norm_kernel(float* __restrict__ degO, float* __restrict__ degI) {
  int i = blockIdx.x * blockDim.x + threadIdx.x;
  if (i >= NN) return;
  degO[i] = rsqrtf(fmaxf(degO[i], 1.0f));
  degI[i] = rsqrtf(fmaxf(degI[i], 1.0f));
}

// ------------------------------------------------------- layer 1 scatter-add
// agg[b,dst,0..7] += feat[b,src,:] * norm_src[src] * ew[b,e]
__global__ void scatter1_kernel(const float* __restrict__ cbg, const float* __restrict__ poi,
                                const float* __restrict__ ew, const int* __restrict__ src,
                                const int* __restrict__ dst, const float* __restrict__ normS,
                                float* __restrict__ agg) {
  int e = blockIdx.x * blockDim.x + threadIdx.x;
  if (e >= NE) return;
  int b = blockIdx.y;
  int s = src[e], d = dst[e];
  float w = ew[(size_t)b * NE + e] * normS[s];
  const float4* f = (s < CBGN)
      ? (const float4*)&cbg[((size_t)b * CBGN + s) * 8]
      : (const float4*)&poi[((size_t)b * POIN + (s - CBGN)) * 8];
  float4 f0 = f[0], f1 = f[1];
  float* o = &agg[((size_t)b * NN + d) * 8];
  atomicAdd(&o[0], f0.x * w); atomicAdd(&o[1], f0.y * w);
  atomicAdd(&o[2], f0.z * w); atomicAdd(&o[3], f0.w * w);
  atomicAdd(&o[4], f1.x * w); atomicAdd(&o[5], f1.y * w);
  atomicAdd(&o[6], f1.z * w); atomicAdd(&o[7], f1.w * w);
}

// ------------------------------------------------------- layer 2 scatter-add
// agg[b,dst,0..15] += h1s[b,src,:] * ew[b,e]   (norm_src already folded in h1s)
__global__ void scatter2_kernel(const float* __restrict__ h1s, const float* __restrict__ ew,
                                const int* __restrict__ src, const int* __restrict__ dst,
                                float* __restrict__ agg) {
  int e = blockIdx.x * blockDim.x + threadIdx.x;
  if (e >= NE) return;
  int b = blockIdx.y;
  int s = src[e], d = dst[e];
  float w = ew[(size_t)b * NE + e];
  const float4* f = (const float4*)&h1s[((size_t)b * NN + s) * 16];
  float4 f0 = f[0], f1 = f[1], f2 = f[2], f3 = f[3];
  float* o = &agg[((size_t)b * NN + d) * 16];
  atomicAdd(&o[0],  f0.x * w); atomicAdd(&o[1],  f0.y * w);
  atomicAdd(&o[2],  f0.z * w); atomicAdd(&o[3],  f0.w * w);
  atomicAdd(&o[4],  f1.x * w); atomicAdd(&o[5],  f1.y * w);
  atomicAdd(&o[6],  f1.z * w); atomicAdd(&o[7],  f1.w * w);
  atomicAdd(&o[8],  f2.x * w); atomicAdd(&o[9],  f2.y * w);
  atomicAdd(&o[10], f2.z * w); atomicAdd(&o[11], f2.w * w);
  atomicAdd(&o[12], f3.x * w); atomicAdd(&o[13], f3.y * w);
  atomicAdd(&o[14], f3.z * w); atomicAdd(&o[15], f3.w * w);
}

// --------------------------------------------------------- dense 1 via WMMA
// h1s[b,n,0..15] = leaky_relu((agg1[b,n,:]*norm_dst[n]) @ W1 + b1) * norm_src[n]
// One wave32 per 16-node tile. V_WMMA_F32_16X16X4_F32, K=8 in two chunks.
// A layout (f32 16x4): lane L<16 -> row L, {K=kb+0,kb+1}; lanes 16..31 -> {K=kb+2,kb+3}.
// B layout (f32 4x16): lane L<16 -> col L, {K=kb+0,kb+1}; lanes 16..31 -> {K=kb+2,kb+3}.
// C/D: c[r] = D[r + 8*(lane>>4)][lane&15].
__global__ void __launch_bounds__(256)
dense1_kernel(const float* __restrict__ agg, const float* __restrict__ normD,
              const float* __restrict__ normS, const float* __restrict__ W1,
              const float* __restrict__ b1, float* __restrict__ h1s) {
  int wave = (int)((blockIdx.x * blockDim.x + threadIdx.x) >> 5);
  int lane = threadIdx.x & 31;
  const int tilesPerB = NN / 16;           // 3125
  int b = wave / tilesPerB;
  int t = wave % tilesPerB;
  if (b >= NB) return;
  int m0 = t * 16;
  int rc = lane & 15;                      // row of A / col of B,D
  int half = lane >> 4;
  float nd = normD[m0 + rc];
  const float* arow = &agg[((size_t)b * NN + m0 + rc) * 8];

  v8f c = {0.f, 0.f, 0.f, 0.f, 0.f, 0.f, 0.f, 0.f};
#pragma unroll
  for (int kb = 0; kb < 8; kb += 4) {
    int k0 = kb + 2 * half;
    v2f a, w;
    a.x = arow[k0 + 0] * nd;
    a.y = arow[k0 + 1] * nd;
    w.x = W1[(k0 + 0) * 16 + rc];
    w.y = W1[(k0 + 1) * 16 + rc];
    c = __builtin_amdgcn_wmma_f32_16x16x4_f32(false, a, false, w, (short)0, c,
                                              false, false);
  }
  float bias = b1[rc];
#pragma unroll
  for (int r = 0; r < 8; ++r) {
    int node = m0 + r + 8 * half;
    float v = c[r] + bias;
    v = v > 0.0f ? v : 0.01f * v;          // leaky_relu, slope 0.01
    v *= normS[node];                      // pre-fold next layer's src norm
    h1s[((size_t)b * NN + node) * 16 + rc] = v;
  }
}

// --------------------------------------------------------- dense 2 via WMMA
// out[b,n,0..7] = (agg2[b,n,:]*norm_dst[n]) @ W2 + b2   for n < CBGN
// K=16 in four chunks; B matrix = W2 zero-padded to 16 columns.
__global__ void __launch_bounds__(256)
dense2_kernel(const float* __restrict__ agg, const float* __restrict__ normD,
              const float* __restrict__ W2, const float* __restrict__ b2,
              float* __restrict__ out) {
  int wave = (int)((blockIdx.x * blockDim.x + threadIdx.x) >> 5);
  int lane = threadIdx.x & 31;
  const int tilesPerB = CBGN / 16;         // 1875
  int b = wave / tilesPerB;
  int t = wave % tilesPerB;
  if (b >= NB) return;
  int m0 = t * 16;
  int rc = lane & 15;
  int half = lane >> 4;
  float nd = normD[m0 + rc];
  const float* arow = &agg[((size_t)b * NN + m0 + rc) * 16];
  int cw = rc & 7;
  float colmask = (rc < 8) ? 1.0f : 0.0f;  // zero-pad cols 8..15, no divergence

  v8f c = {0.f, 0.f, 0.f, 0.f, 0.f, 0.f, 0.f, 0.f};
#pragma unroll
  for (int kb = 0; kb < 16; kb += 4) {
    int k0 = kb + 2 * half;
    v2f a, w;
    a.x = arow[k0 + 0] * nd;
    a.y = arow[k0 + 1] * nd;
    w.x = W2[(k0 + 0) * 8 + cw] * colmask;
    w.y = W2[(k0 + 1) * 8 + cw] * colmask;
    c = __builtin_amdgcn_wmma_f32_16x16x4_f32(false, a, false, w, (short)0, c,
                                              false, false);
  }
  if (rc < 8) {
    float bias = b2[rc];
#pragma unroll
    for (int r = 0; r < 8; ++r) {
      int node = m0 + r + 8 * half;
      out[((size_t)b * CBGN + node) * 8 + rc] = c[r] + bias;
    }
  }
}

// --------------------------------------------------------------------------
extern "C" void kernel_launch(void* const* d_in, const int* in_sizes, int n_in,
                              void* d_out, int out_size, void* d_ws, size_t ws_size,
                              hipStream_t stream) {
  (void)in_sizes; (void)n_in; (void)out_size; (void)ws_size;
  const float* ew  = (const float*)d_in[0];   // [B,E]
  const float* cbg = (const float*)d_in[1];   // [B,CBG,8]
  const float* poi = (const float*)d_in[2];   // [B,POI,8]
  const float* W1  = (const float*)d_in[3];   // [8,16]
  const float* b1  = (const float*)d_in[4];   // [16]
  const float* W2  = (const float*)d_in[5];   // [16,8]
  const float* b2  = (const float*)d_in[6];   // [8]
  const int*   src = (const int*)d_in[7];     // [E]
  const int*   dst = (const int*)d_in[8];     // [E]
  float* out = (float*)d_out;                 // [B,CBG,8]

  float* ws    = (float*)d_ws;
  float* normS = ws;                                   // N
  float* normD = ws + NN;                              // N
  float* agg   = ws + 2 * (size_t)NN;                  // B*N*16 (reused L1/L2)
  float* h1s   = agg + (size_t)NB * NN * 16;           // B*N*16

  // 1) zero degree accumulators + layer-1 agg region
  long long z1 = 2LL * NN + (long long)NB * NN * 16;
  zero_f32<<<(int)((z1 + 255) / 256), 256, 0, stream>>>(normS, z1);

  // 2) degrees -> norms
  degree_kernel<<<(NE + 255) / 256, 256, 0, stream>>>(src, dst, normS, normD);
  norm_kernel<<<(NN + 255) / 256, 256, 0, stream>>>(normS, normD);

  // 3) layer-1 message scatter
  dim3 gE((NE + 255) / 256, NB);
  scatter1_kernel<<<gE, 256, 0, stream>>>(cbg, poi, ew, src, dst, normS, agg);

  // 4) layer-1 dense (WMMA) + activation + norm_src fold
  dense1_kernel<<<(NB * (NN / 16)) / 8, 256, 0, stream>>>(agg, normD, normS, W1, b1, h1s);

  // 5) re-zero agg for layer 2
  long long z2 = (long long)NB * NN * 16;
  zero_f32<<<(int)((z2 + 255) / 256), 256, 0, stream>>>(agg, z2);

  // 6) layer-2 message scatter
  scatter2_kernel<<<gE, 256, 0, stream>>>(h1s, ew, src, dst, agg);

  // 7) layer-2 dense (WMMA) -> output (CBG nodes only)
  dense2_kernel<<<(NB * (CBGN / 16)) / 8, 256, 0, stream>>>(agg, normD, W2, b2, out);
}